// RecallLoss_41369124995560
// MI455X (gfx1250) — compile-verified
//
#include <hip/hip_runtime.h>
#include <hip/hip_bf16.h>

#define SMOOTH 1e-05f

// Problem constants (from reference setup_inputs)
#define RN 8
#define RC 21
#define RHW (512 * 512)          // pixels per image
#define RNC (RN * RC)            // 168 histogram bins
#define RP (RN * RHW)            // 2,097,152 total pixels

typedef __attribute__((ext_vector_type(2))) float v2f;
typedef __attribute__((ext_vector_type(8))) float v8f;

// ---------------------------------------------------------------------------
// Kernel 0: zero the tp/tt accumulators in workspace (2*168 u32)
// ---------------------------------------------------------------------------
__global__ __launch_bounds__(384) void recall_zero(unsigned* __restrict__ ws) {
    int i = threadIdx.x;
    if (i < 2 * RNC) ws[i] = 0u;
}

// ---------------------------------------------------------------------------
// Kernel 1: streaming argmax + histogram.
//   input: NCHW fp32. For fixed c, lanes load consecutive pixels -> coalesced.
//   Non-temporal loads: 176MB single-use stream, keep L2 clean.
//   LDS histograms (ds_add atomics) flushed once per block via global atomics.
// ---------------------------------------------------------------------------
__global__ __launch_bounds__(256) void recall_main(const float* __restrict__ inp,
                                                   const int*   __restrict__ tgt_lo, // int64 viewed as int pairs
                                                   unsigned*    __restrict__ g_tp,
                                                   unsigned*    __restrict__ g_tt) {
    __shared__ unsigned s_tp[RNC];
    __shared__ unsigned s_tt[RNC];
    for (int i = threadIdx.x; i < RNC; i += blockDim.x) { s_tp[i] = 0u; s_tt[i] = 0u; }
    __syncthreads();

    const int stride = gridDim.x * blockDim.x;
    for (int idx = blockIdx.x * blockDim.x + threadIdx.x; idx < RP; idx += stride) {
        const int n = idx >> 18;            // idx / (512*512)
        const int l = idx & (RHW - 1);      // idx % (512*512)
        const float* base = inp + (size_t)n * RC * RHW + l;

        // Prefetch next grid-stride chunk (speculative; safe if OOB)
        __builtin_prefetch(base + (size_t)stride, 0, 0);

        // argmax over channels; strict '>' keeps first-max semantics like jnp.argmax
        float best = __builtin_nontemporal_load(base);
        int bc = 0;
#pragma unroll
        for (int c = 1; c < RC; ++c) {
            float v = __builtin_nontemporal_load(base + (size_t)c * RHW);
            if (v > best) { best = v; bc = c; }
        }

        const int t = tgt_lo[2 * idx];      // low dword of little-endian int64 (values 0..20)
        atomicAdd(&s_tt[n * RC + t], 1u);
        if (bc == t) atomicAdd(&s_tp[n * RC + t], 1u);
    }

    __syncthreads();
    for (int i = threadIdx.x; i < RNC; i += blockDim.x) {
        unsigned vtt = s_tt[i];
        unsigned vtp = s_tp[i];
        if (vtt) atomicAdd(&g_tt[i], vtt);
        if (vtp) atomicAdd(&g_tp[i], vtp);
    }
}

// ---------------------------------------------------------------------------
// Kernel 2: finalize with V_WMMA_F32_16X16X4_F32.
//   recall[i] = (tp[i]+eps)/(tt[i]+eps), i in [0,168), padded with zeros to
//   3 tiles of 64 (A = 16x4 f32). B = ones, C accumulates across tiles.
//   D[M][N] = S_M for every N, so:
//     lanes 0..15 : sum of their 8 acc VGPRs = S_0+..+S_7
//     lanes 16..31: sum of their 8 acc VGPRs = S_8+..+S_15
//   One shfl_xor(16) -> exact total sum in all lanes.
//   Single wave of 32 threads => EXEC all ones (WMMA requirement).
// ---------------------------------------------------------------------------
__global__ __launch_bounds__(32) void recall_finalize(const unsigned* __restrict__ g_tp,
                                                      const unsigned* __restrict__ g_tt,
                                                      float* __restrict__ out) {
    const int lane = threadIdx.x;   // 0..31

    v8f acc = {0.f, 0.f, 0.f, 0.f, 0.f, 0.f, 0.f, 0.f};
    v2f bones;
    bones.x = 1.0f;
    bones.y = 1.0f;

#pragma unroll
    for (int t = 0; t < 3; ++t) {
        const int i0 = t * 64 + lane * 2;
        const int i1 = i0 + 1;
        v2f a;
        a.x = (i0 < RNC) ? (((float)g_tp[i0] + SMOOTH) / ((float)g_tt[i0] + SMOOTH)) : 0.0f;
        a.y = (i1 < RNC) ? (((float)g_tp[i1] + SMOOTH) / ((float)g_tt[i1] + SMOOTH)) : 0.0f;
        // 8 args: (neg_a, A, neg_b, B, c_mod, C, reuse_a, reuse_b)
        acc = __builtin_amdgcn_wmma_f32_16x16x4_f32(false, a, false, bones,
                                                    (short)0, acc, false, false);
    }

    float local = acc[0] + acc[1] + acc[2] + acc[3] + acc[4] + acc[5] + acc[6] + acc[7];
    local += __shfl_xor(local, 16, 32);     // total = sum of all 168 recall values

    if (lane == 0) out[0] = 1.0f - local / (float)RNC;
}

// ---------------------------------------------------------------------------
extern "C" void kernel_launch(void* const* d_in, const int* in_sizes, int n_in,
                              void* d_out, int out_size, void* d_ws, size_t ws_size,
                              hipStream_t stream) {
    const float* inp    = (const float*)d_in[0];   // [8,21,512,512] f32
    const int*   tgt_lo = (const int*)d_in[1];     // [8,512,512] int64 -> int pairs

    unsigned* g_tp = (unsigned*)d_ws;              // 168 u32
    unsigned* g_tt = g_tp + RNC;                   // 168 u32

    recall_zero<<<1, 384, 0, stream>>>(g_tp);
    recall_main<<<2048, 256, 0, stream>>>(inp, tgt_lo, g_tp, g_tt);
    recall_finalize<<<1, 32, 0, stream>>>(g_tp, g_tt, (float*)d_out);
}